// GAT_5686536700271
// MI455X (gfx1250) — compile-verified
//
#include <hip/hip_runtime.h>
#include <hip/hip_bf16.h>

// ---------------------------------------------------------------------------
// GAT 2-layer + FC for MI455X (gfx1250, wave32).
//   - Dense transforms via V_WMMA_F32_16X16X4_F32 (exact fp32 matrix path)
//   - Edge passes designed so all node arrays live in the 192MB L2
//   - segment_max via ordered-uint atomicMax; memset(0) == -inf init
// ---------------------------------------------------------------------------

typedef float v2f __attribute__((ext_vector_type(2)));
typedef float v8f __attribute__((ext_vector_type(8)));

#define NEG_SLOPE 0.2f

// ---------------------- WMMA GEMM: H = X[n_rows x K] @ W[KxNOUT] -------------------
// Block = 128 threads = 4 waves. Each wave owns one 16x16 output tile.
// CT = NOUT/16 column tiles, ROWT = 4/CT row tiles per block.
// A fragment (16x4 f32): lanes 0-15 hold K= k0+0 / k0+1 for rows M=lane,
//                        lanes 16-31 hold K= k0+2 / k0+3 for rows M=lane-16.
// B fragment (4x16 f32): mirrored: v0 = W[k0+khalf][n], v1 = W[k0+khalf+1][n].
// D (16x16 f32, 8 VGPRs): VGPR r -> M=r (lanes 0-15), M=r+8 (lanes 16-31).
template <int K, int NOUT, int ROWT>
__global__ __launch_bounds__(128) void gemm_wmma_f32(
    const float* __restrict__ X, const float* __restrict__ W,
    float* __restrict__ H, int n_rows) {
  constexpr int KP = K + 1;  // pad: stride K+1 spreads lanes across LDS banks
  __shared__ float sX[ROWT * 16 * KP];
  __shared__ float sW[K * NOUT];

  const int row_base = blockIdx.x * (ROWT * 16);

  for (int idx = threadIdx.x; idx < ROWT * 16 * K; idx += 128) {
    const int r = idx / K, c = idx % K;
    const int gr = row_base + r;
    sX[r * KP + c] = (gr < n_rows) ? X[(long long)gr * K + c] : 0.0f;
  }
  for (int idx = threadIdx.x; idx < K * NOUT; idx += 128) sW[idx] = W[idx];
  __syncthreads();

  const int lane = threadIdx.x & 31;
  const int wave = threadIdx.x >> 5;
  constexpr int CT = NOUT / 16;
  const int ct = wave % CT;
  const int rt = wave / CT;
  const int m = lane & 15;
  const int khalf = (lane >> 4) * 2;  // 0 for lanes 0-15, 2 for lanes 16-31

  v8f c = {};
  const float* xrow = &sX[(rt * 16 + m) * KP];
#pragma unroll
  for (int k0 = 0; k0 < K; k0 += 4) {
    v2f a, b;
    a.x = xrow[k0 + khalf];
    a.y = xrow[k0 + khalf + 1];
    b.x = sW[(k0 + khalf) * NOUT + ct * 16 + m];
    b.y = sW[(k0 + khalf + 1) * NOUT + ct * 16 + m];
    // (neg_a, A, neg_b, B, c_mod, C, reuse_a, reuse_b)
    c = __builtin_amdgcn_wmma_f32_16x16x4_f32(false, a, false, b, (short)0, c,
                                              false, false);
  }

  const int gm_base = row_base + rt * 16 + (lane >> 4) * 8;
  const int n = ct * 16 + (lane & 15);
#pragma unroll
  for (int r = 0; r < 8; ++r) {
    const int gr = gm_base + r;
    if (gr < n_rows) H[(long long)gr * NOUT + n] = c[r];
  }
}

// ---------------------- per-node attention dot products ---------------------
template <int H>
__global__ __launch_bounds__(256) void node_dots(
    const float* __restrict__ h, const float* __restrict__ a_src,
    const float* __restrict__ a_dst, float* __restrict__ out_s,
    float* __restrict__ out_d, int n) {
  const int i = blockIdx.x * blockDim.x + threadIdx.x;
  if (i >= n) return;
  float s = 0.0f, d = 0.0f;
#pragma unroll
  for (int f = 0; f < H; ++f) {
    const float v = h[(long long)i * H + f];
    s += v * a_src[f];
    d += v * a_dst[f];
  }
  out_s[i] = s;
  out_d[i] = d;
}

// ---------------------- ordered-uint float max trick ------------------------
__device__ __forceinline__ unsigned f2ord(float f) {
  const unsigned b = __float_as_uint(f);
  return (b & 0x80000000u) ? ~b : (b | 0x80000000u);
}
__device__ __forceinline__ float ord2f(unsigned u) {
  const unsigned b = (u & 0x80000000u) ? (u ^ 0x80000000u) : ~u;
  return __uint_as_float(b);
}

__device__ __forceinline__ void edge_pair(const int* __restrict__ ei, int e,
                                          int E, int& s, int& d) {
  if (e < E) {
    s = ei[e];
    d = ei[E + e];
  } else {
    s = d = e - E;  // self loop appended after the E real edges
  }
}

// ------------ pass 1: score + segment max over destinations ----------------
__global__ __launch_bounds__(256) void edge_max_k(
    const int* __restrict__ ei, const float* __restrict__ as,
    const float* __restrict__ ad, unsigned* __restrict__ mo, int E, int NE) {
  const int e = blockIdx.x * blockDim.x + threadIdx.x;
  if (e >= NE) return;
  if (e + 4096 < E) __builtin_prefetch(ei + e + 4096, 0, 1);
  int s, d;
  edge_pair(ei, e, E, s, d);
  float sc = as[s] + ad[d];
  sc = sc > 0.0f ? sc : sc * NEG_SLOPE;
  atomicMax(&mo[d], f2ord(sc));
}

// ------------ pass 2: p = exp(e - m[dst]); segment sum ---------------------
__global__ __launch_bounds__(256) void edge_sum_k(
    const int* __restrict__ ei, const float* __restrict__ as,
    const float* __restrict__ ad, const unsigned* __restrict__ mo,
    float* __restrict__ ssum, float* __restrict__ pbuf, int E, int NE) {
  const int e = blockIdx.x * blockDim.x + threadIdx.x;
  if (e >= NE) return;
  if (e + 4096 < E) __builtin_prefetch(ei + e + 4096, 0, 1);
  int s, d;
  edge_pair(ei, e, E, s, d);
  float sc = as[s] + ad[d];
  sc = sc > 0.0f ? sc : sc * NEG_SLOPE;
  const float p = __expf(sc - ord2f(mo[d]));
  pbuf[e] = p;
  atomicAdd(&ssum[d], p);
}

// ------------ pass 3: acc[dst] += alpha * h[src] ---------------------------
template <int H>
__global__ __launch_bounds__(256) void edge_scatter_k(
    const int* __restrict__ ei, const float* __restrict__ h,
    const float* __restrict__ pbuf, const float* __restrict__ ssum,
    float* __restrict__ acc, int E, int NE) {
  const unsigned t = blockIdx.x * blockDim.x + threadIdx.x;
  const unsigned total = (unsigned)NE * (unsigned)H;
  if (t >= total) return;
  const int e = t / H;
  const int f = t % H;
  int s, d;
  edge_pair(ei, e, E, s, d);
  const float alpha = pbuf[e] / (ssum[d] + 1e-16f);
  atomicAdd(&acc[(long long)d * H + f], alpha * h[(long long)s * H + f]);
}

// ------------ bias + relu (in place) ---------------------------------------
__global__ __launch_bounds__(256) void bias_relu_k(float* __restrict__ acc,
                                                   const float* __restrict__ b,
                                                   int total, int H) {
  const int t = blockIdx.x * blockDim.x + threadIdx.x;
  if (t >= total) return;
  acc[t] = fmaxf(acc[t] + b[t % H], 0.0f);
}

// ------------ FC head: logits = emb @ Wfc + bfc  (32 -> 40) ----------------
__global__ __launch_bounds__(256) void fc_k(const float* __restrict__ emb,
                                            const float* __restrict__ Wfc,
                                            const float* __restrict__ bfc,
                                            float* __restrict__ out, int n) {
  const int t = blockIdx.x * blockDim.x + threadIdx.x;
  if (t >= n * 40) return;
  const int i = t / 40, o = t % 40;
  float acc = bfc[o];
#pragma unroll
  for (int k = 0; k < 32; ++k) acc += emb[i * 32 + k] * Wfc[k * 40 + o];
  out[t] = acc;
}

// ---------------------------------------------------------------------------
extern "C" void kernel_launch(void* const* d_in, const int* in_sizes, int n_in,
                              void* d_out, int out_size, void* d_ws,
                              size_t ws_size, hipStream_t stream) {
  const float* x      = (const float*)d_in[0];
  const int*   ei     = (const int*)d_in[1];
  const float* W1     = (const float*)d_in[2];
  const float* a_src1 = (const float*)d_in[3];
  const float* a_dst1 = (const float*)d_in[4];
  const float* b1     = (const float*)d_in[5];
  const float* W2     = (const float*)d_in[6];
  const float* a_src2 = (const float*)d_in[7];
  const float* a_dst2 = (const float*)d_in[8];
  const float* b2     = (const float*)d_in[9];
  const float* Wfc    = (const float*)d_in[10];
  const float* bfc    = (const float*)d_in[11];

  const int N  = in_sizes[0] / 256;
  const int E  = in_sizes[1] / 2;
  const int NE = E + N;  // real edges + self loops

  // workspace carve-up (floats)
  float* ws    = (float*)d_ws;
  float* h1    = ws;                          // N*64
  float* acc1  = h1 + (size_t)N * 64;         // N*64
  float* h2    = acc1 + (size_t)N * 64;       // N*32
  float* asv   = h2 + (size_t)N * 32;         // N
  float* adv   = asv + N;                     // N
  unsigned* mo = (unsigned*)(adv + N);        // N
  float* ssum  = (float*)(mo + N);            // N
  float* pbuf  = ssum + N;                    // NE

  float* emb    = (float*)d_out;              // N*32
  float* logits = emb + (size_t)N * 32;       // N*40

  const int B = 256;
  const int gN   = (N + B - 1) / B;
  const int gNE  = (NE + B - 1) / B;

  // =============================== layer 1 =================================
  gemm_wmma_f32<256, 64, 1><<<(N + 15) / 16, 128, 0, stream>>>(x, W1, h1, N);
  node_dots<64><<<gN, B, 0, stream>>>(h1, a_src1, a_dst1, asv, adv, N);

  hipMemsetAsync(mo, 0, (size_t)N * 4, stream);       // ordered(0) == -max
  hipMemsetAsync(ssum, 0, (size_t)N * 4, stream);
  hipMemsetAsync(acc1, 0, (size_t)N * 64 * 4, stream);

  edge_max_k<<<gNE, B, 0, stream>>>(ei, asv, adv, mo, E, NE);
  edge_sum_k<<<gNE, B, 0, stream>>>(ei, asv, adv, mo, ssum, pbuf, E, NE);
  {
    const long long tot = (long long)NE * 64;
    edge_scatter_k<64><<<(int)((tot + B - 1) / B), B, 0, stream>>>(
        ei, h1, pbuf, ssum, acc1, E, NE);
  }
  bias_relu_k<<<(N * 64 + B - 1) / B, B, 0, stream>>>(acc1, b1, N * 64, 64);

  // =============================== layer 2 =================================
  gemm_wmma_f32<64, 32, 2><<<(N + 31) / 32, 128, 0, stream>>>(acc1, W2, h2, N);
  node_dots<32><<<gN, B, 0, stream>>>(h2, a_src2, a_dst2, asv, adv, N);

  hipMemsetAsync(mo, 0, (size_t)N * 4, stream);
  hipMemsetAsync(ssum, 0, (size_t)N * 4, stream);
  hipMemsetAsync(emb, 0, (size_t)N * 32 * 4, stream);

  edge_max_k<<<gNE, B, 0, stream>>>(ei, asv, adv, mo, E, NE);
  edge_sum_k<<<gNE, B, 0, stream>>>(ei, asv, adv, mo, ssum, pbuf, E, NE);
  {
    const long long tot = (long long)NE * 32;
    edge_scatter_k<32><<<(int)((tot + B - 1) / B), B, 0, stream>>>(
        ei, h2, pbuf, ssum, emb, E, NE);
  }
  bias_relu_k<<<(N * 32 + B - 1) / B, B, 0, stream>>>(emb, b2, N * 32, 32);

  // =============================== FC head =================================
  fc_k<<<(N * 40 + B - 1) / B, B, 0, stream>>>(emb, Wfc, bfc, logits, N);
}